// Representation_Projector_8538394985041
// MI455X (gfx1250) — compile-verified
//
#include <hip/hip_runtime.h>

#define B_  2
#define L_  512
#define S_  8
#define CM_ 256
#define CZ_ 128
#define IN_ 21
#define NB_ 65

typedef __attribute__((ext_vector_type(2))) float v2f;
typedef __attribute__((ext_vector_type(4))) float v4f;
typedef __attribute__((ext_vector_type(8))) float v8f;

// ---------------------------------------------------------------------------
// Kernel 1: li = seqs@W1+b1, lj = seqs@W2+b2  via V_WMMA_F32_16X16X4_F32.
// K = 21: five fully in-bounds K=4 steps (branch-free 64-bit A loads), plus
// one masked tail step (unconditional loads + v_cndmask, no EXEC divergence).
// block = (32,4): 4 waves, each owns one 16x16 tile. A tile feeds both accs.
// ---------------------------------------------------------------------------
__global__ void __launch_bounds__(128)
proj_li_lj_kernel(const float* __restrict__ seqs,
                  const float* __restrict__ W1, const float* __restrict__ b1,
                  const float* __restrict__ W2, const float* __restrict__ b2,
                  float* __restrict__ li, float* __restrict__ lj) {
  const int lane = threadIdx.x;
  const int half = lane >> 4;                       // 0: K pair lo, 1: K pair hi
  const int sub  = lane & 15;                       // M for A-frag, N for B-frag
  const int n0   = blockIdx.x * 16;                 // CZ tile
  const int m0   = (blockIdx.y * 4 + threadIdx.y) * 16;  // (B*L) row tile
  const int col  = n0 + sub;

  const float* arow = seqs + (size_t)(m0 + sub) * IN_;

  v8f acc1 = {};
  v8f acc2 = {};

#pragma unroll
  for (int kk = 0; kk < 20; kk += 4) {              // in-bounds: ka+1 <= 19 < 21
    const int ka = kk + 2 * half;
    v2f a  = { arow[ka], arow[ka + 1] };
    v2f f1 = { W1[(size_t)ka * CZ_ + col], W1[(size_t)(ka + 1) * CZ_ + col] };
    v2f f2 = { W2[(size_t)ka * CZ_ + col], W2[(size_t)(ka + 1) * CZ_ + col] };
    acc1 = __builtin_amdgcn_wmma_f32_16x16x4_f32(false, a, false, f1,
                                                 (short)0, acc1, false, false);
    acc2 = __builtin_amdgcn_wmma_f32_16x16x4_f32(false, a, false, f2,
                                                 (short)0, acc2, false, false);
  }

  // Tail kk=20: only k=20 is real (lanes 0-15); k=21..23 are zero padding.
  {
    const float av  = arow[20];                     // unconditional, valid addr
    const float w1v = W1[20 * CZ_ + col];
    const float w2v = W2[20 * CZ_ + col];
    v2f a  = { half ? 0.0f : av,  0.0f };           // lane-select, no branches
    v2f f1 = { half ? 0.0f : w1v, 0.0f };
    v2f f2 = { half ? 0.0f : w2v, 0.0f };
    acc1 = __builtin_amdgcn_wmma_f32_16x16x4_f32(false, a, false, f1,
                                                 (short)0, acc1, false, false);
    acc2 = __builtin_amdgcn_wmma_f32_16x16x4_f32(false, a, false, f2,
                                                 (short)0, acc2, false, false);
  }

  const float bias1 = b1[col];
  const float bias2 = b2[col];
#pragma unroll
  for (int r = 0; r < 8; ++r) {
    const int row = m0 + r + 8 * half;
    li[(size_t)row * CZ_ + col] = acc1[r] + bias1;
    lj[(size_t)row * CZ_ + col] = acc2[r] + bias2;
  }
}

// ---------------------------------------------------------------------------
// Kernel 2: msa[b,s,l,m] = evos[b] @ Wp[s] + bp[s]  (same WMMA structure).
// block = (32,4): 4 waves over consecutive L tiles.
// ---------------------------------------------------------------------------
__global__ void __launch_bounds__(128)
msa_proj_kernel(const float* __restrict__ evos,
                const float* __restrict__ Wp, const float* __restrict__ bp,
                float* __restrict__ msa) {
  const int lane = threadIdx.x;
  const int half = lane >> 4;
  const int sub  = lane & 15;
  const int n0   = blockIdx.x * 16;                 // CM tile
  const int l0   = (blockIdx.y * 4 + threadIdx.y) * 16;  // L tile
  const int bs   = blockIdx.z;
  const int b    = bs / S_;
  const int s    = bs % S_;
  const int col  = n0 + sub;

  const float* arow = evos + ((size_t)b * L_ + l0 + sub) * IN_;
  const float* W    = Wp + (size_t)s * IN_ * CM_;

  v8f acc = {};

#pragma unroll
  for (int kk = 0; kk < 20; kk += 4) {
    const int ka = kk + 2 * half;
    v2f a = { arow[ka], arow[ka + 1] };
    v2f f = { W[(size_t)ka * CM_ + col], W[(size_t)(ka + 1) * CM_ + col] };
    acc = __builtin_amdgcn_wmma_f32_16x16x4_f32(false, a, false, f,
                                                (short)0, acc, false, false);
  }
  {
    const float av = arow[20];
    const float wv = W[20 * CM_ + col];
    v2f a = { half ? 0.0f : av, 0.0f };
    v2f f = { half ? 0.0f : wv, 0.0f };
    acc = __builtin_amdgcn_wmma_f32_16x16x4_f32(false, a, false, f,
                                                (short)0, acc, false, false);
  }

  const float bias = bp[(size_t)s * CM_ + col];
  float* outbase = msa + (((size_t)b * S_ + s) * L_) * CM_;
#pragma unroll
  for (int r = 0; r < 8; ++r) {
    const int row = l0 + r + 8 * half;
    outbase[(size_t)row * CM_ + col] = acc[r] + bias;
  }
}

// ---------------------------------------------------------------------------
// Kernel 3: prw[b,x,y,c] = li[b,y,c] + lj[b,x,c] + Wr[clamp(y-x,-32,32)+32,c]+br[c]
// Store-bandwidth kernel (256 MB out -> ~11 us floor at 23.3 TB/s).
// relpos table (65x128 fp32 = 33 KB, bias folded in) in LDS; li/lj stay
// L2-resident; output via non-temporal b128 stores to keep L2 clean.
// grid = (L/256, L, B), block = 256. Each block: one (b,x), 256 y rows
// (128 KB out per block vs 33 KB table fill -> fill is ~25% extra L2 reads).
// ---------------------------------------------------------------------------
__global__ void __launch_bounds__(256)
prw_kernel(const float* __restrict__ li, const float* __restrict__ lj,
           const float* __restrict__ Wr, const float* __restrict__ br,
           float* __restrict__ out) {
  __shared__ float WrS[NB_ * CZ_];

  const int tid = threadIdx.x;
  for (int i = tid; i < NB_ * CZ_; i += 256)
    WrS[i] = Wr[i] + br[i & (CZ_ - 1)];             // fold bias into table
  __syncthreads();

  const int x = blockIdx.y;
  const int b = blockIdx.z;
  const int c     = (tid & 31) * 4;                 // float4 group in CZ=128
  const int ysub  = tid >> 5;                       // 0..7
  const int ybase = blockIdx.x * 256;

  const v4f lj4 = *reinterpret_cast<const v4f*>(
      lj + ((size_t)b * L_ + x) * CZ_ + c);

  const float* libase = li + (size_t)b * L_ * CZ_ + c;
  float* orow = out + ((((size_t)b * L_ + x) * L_) * CZ_) + c;

#pragma unroll 8
  for (int yi = 0; yi < 32; ++yi) {
    const int y = ybase + yi * 8 + ysub;
    int d = y - x;
    d = d < -32 ? -32 : (d > 32 ? 32 : d);
    d += 32;
    const v4f li4 = *reinterpret_cast<const v4f*>(libase + (size_t)y * CZ_);
    const v4f w4  = *reinterpret_cast<const v4f*>(WrS + d * CZ_);
    v4f o = li4 + lj4 + w4;
    __builtin_nontemporal_store(o, reinterpret_cast<v4f*>(orow + (size_t)y * CZ_));
  }
}

// ---------------------------------------------------------------------------
extern "C" void kernel_launch(void* const* d_in, const int* in_sizes, int n_in,
                              void* d_out, int out_size, void* d_ws, size_t ws_size,
                              hipStream_t stream) {
  const float* seqs = (const float*)d_in[0];
  const float* evos = (const float*)d_in[1];
  const float* W1   = (const float*)d_in[2];
  const float* b1   = (const float*)d_in[3];
  const float* W2   = (const float*)d_in[4];
  const float* b2   = (const float*)d_in[5];
  const float* Wp   = (const float*)d_in[6];
  const float* bp   = (const float*)d_in[7];
  const float* Wr   = (const float*)d_in[8];
  const float* br   = (const float*)d_in[9];

  float* prw = (float*)d_out;                                   // [B,L,L,CZ]
  float* msa = prw + (size_t)B_ * L_ * L_ * CZ_;                // [B,S,L,CM]

  float* li = (float*)d_ws;                                     // [B,L,CZ]
  float* lj = li + (size_t)B_ * L_ * CZ_;                       // [B,L,CZ]

  // li / lj projections: 8 x 16 blocks x 4 waves = 512 16x16 tiles
  proj_li_lj_kernel<<<dim3(CZ_ / 16, (B_ * L_) / 64), dim3(32, 4), 0, stream>>>(
      seqs, W1, b1, W2, b2, li, lj);

  // msa projections: 16 x 8 x 16 blocks x 4 waves = 8192 16x16 tiles
  msa_proj_kernel<<<dim3(CM_ / 16, L_ / 64, B_ * S_), dim3(32, 4), 0, stream>>>(
      evos, Wp, bp, msa);

  // pair representation: bandwidth-bound outer sum + relpos lookup
  prw_kernel<<<dim3(L_ / 256, L_, B_), 256, 0, stream>>>(li, lj, Wr, br, prw);
}